// GaitPINN_42949673584
// MI455X (gfx1250) — compile-verified
//
#include <hip/hip_runtime.h>
#include <hip/hip_bf16.h>

typedef float v2f __attribute__((ext_vector_type(2)));
typedef float v8f __attribute__((ext_vector_type(8)));

#define B_TOTAL 16384
#define T_STEPS 256

// ---------------------------------------------------------------------------
// Encoder: per 16-row tile, one wave32.  9 -> 64 -> 64 (ReLU, fp32 WMMA) -> 2.
// Uses V_WMMA_F32_16X16X4_F32 (CDNA5 fp32 matrix path) for the two big layers.
//
// Operand layouts (ISA 7.12.2, fp32, wave32):
//   A (16x4):  lane L holds row M = L&15; VGPR0/.x = K=kbase+(L<16?0:2),
//              VGPR1/.y = K=kbase+(L<16?1:3)
//   B (4x16):  lane L holds col N = L&15; .x = row K=kbase+(L<16?0:2),
//              .y = row K=kbase+(L<16?1:3)
//   C/D (16x16): VGPR r holds M = r + (L<16?0:8), N = L&15
//
// K-padding (9 -> 12) is done branch-free: load from a clamped in-bounds
// address unconditionally, then v_cndmask the value to zero.  This avoids
// the exec-mask save/branch/restore around every padded global_load.
// ---------------------------------------------------------------------------
__global__ __launch_bounds__(32)
void encoder_kernel(const float* __restrict__ x,
                    const float* __restrict__ We1, const float* __restrict__ be1,
                    const float* __restrict__ We2, const float* __restrict__ be2,
                    const float* __restrict__ We3, const float* __restrict__ be3,
                    float* __restrict__ z0_out)
{
    __shared__ float h1[16][68];   // layer-1 activations (padded vs bank conflicts)
    __shared__ float h2[16][68];   // layer-2 activations

    const int lane   = threadIdx.x;       // 0..31
    const int row0   = blockIdx.x * 16;   // first batch row of this tile
    const int m      = lane & 15;         // M row (A) / N col (B,C)
    const int hiHalf = lane >> 4;         // 0: lanes 0-15, 1: lanes 16-31
    const int kOff   = hiHalf * 2;

    const v8f zero8 = {0.f, 0.f, 0.f, 0.f, 0.f, 0.f, 0.f, 0.f};

    // ---------------- Layer 1: (16x9) @ We1(9x64) + be1, ReLU ----------------
    v8f acc[4];
    #pragma unroll
    for (int nt = 0; nt < 4; ++nt) acc[nt] = zero8;

    const size_t xrow = (size_t)(row0 + m) * (T_STEPS * 9);

    #pragma unroll
    for (int kb = 0; kb < 12; kb += 4) {        // K padded 9 -> 12
        const int k0 = kb + kOff;
        const int k1 = k0 + 1;
        const int k0c = (k0 < 9) ? k0 : 8;      // clamped, always in-bounds
        const int k1c = (k1 < 9) ? k1 : 8;

        v2f a;                                   // unconditional loads + select
        a.x = x[xrow + k0c];
        a.y = x[xrow + k1c];
        a.x = (k0 < 9) ? a.x : 0.0f;
        a.y = (k1 < 9) ? a.y : 0.0f;

        #pragma unroll
        for (int nt = 0; nt < 4; ++nt) {
            const int n = nt * 16 + m;
            v2f b;
            b.x = We1[k0c * 64 + n];
            b.y = We1[k1c * 64 + n];
            b.x = (k0 < 9) ? b.x : 0.0f;
            b.y = (k1 < 9) ? b.y : 0.0f;
            acc[nt] = __builtin_amdgcn_wmma_f32_16x16x4_f32(
                false, a, false, b, (short)0, acc[nt], false, false);
        }
    }
    #pragma unroll
    for (int nt = 0; nt < 4; ++nt) {
        const int n = nt * 16 + m;
        #pragma unroll
        for (int r = 0; r < 8; ++r) {
            const float v = acc[nt][r] + be1[n];
            h1[r + hiHalf * 8][n] = v > 0.0f ? v : 0.0f;
        }
    }
    __syncthreads();

    // ---------------- Layer 2: (16x64) @ We2(64x64) + be2, ReLU --------------
    #pragma unroll
    for (int nt = 0; nt < 4; ++nt) acc[nt] = zero8;

    #pragma unroll 4
    for (int kb = 0; kb < 64; kb += 4) {
        const int k0 = kb + kOff;
        v2f a;
        a.x = h1[m][k0];
        a.y = h1[m][k0 + 1];
        #pragma unroll
        for (int nt = 0; nt < 4; ++nt) {
            const int n = nt * 16 + m;
            v2f b;
            b.x = We2[(k0    ) * 64 + n];
            b.y = We2[(k0 + 1) * 64 + n];
            acc[nt] = __builtin_amdgcn_wmma_f32_16x16x4_f32(
                false, a, false, b, (short)0, acc[nt], false, false);
        }
    }
    #pragma unroll
    for (int nt = 0; nt < 4; ++nt) {
        const int n = nt * 16 + m;
        #pragma unroll
        for (int r = 0; r < 8; ++r) {
            const float v = acc[nt][r] + be2[n];
            h2[r + hiHalf * 8][n] = v > 0.0f ? v : 0.0f;
        }
    }
    __syncthreads();

    // ---------------- Layer 3: (16x64) @ We3(64x2) + be3 ---------------------
    // Each lane produces one output: row = m, col = hiHalf.
    float s = be3[hiHalf];
    #pragma unroll 8
    for (int k = 0; k < 64; ++k)
        s = fmaf(h2[m][k], We3[k * 2 + hiHalf], s);
    z0_out[(size_t)(row0 + m) * 2 + hiHalf] = s;
}

// ---------------------------------------------------------------------------
// ODE vector field: vf(z) = tanh(z @ Wo1 + bo1) @ Wo2 + bo2   (2 -> 32 -> 2)
// tanh via native v_exp_f32, overflow-safe: tanh(x) = 1 - 2/(exp(2x)+1)
// ---------------------------------------------------------------------------
__device__ __forceinline__ void vf2(float z0, float z1,
                                    const float* __restrict__ w1,  // 2x32
                                    const float* __restrict__ b1,  // 32
                                    const float* __restrict__ w2,  // 32x2
                                    float b20, float b21,
                                    float& o0, float& o1)
{
    float a0 = b20, a1 = b21;
    #pragma unroll
    for (int j = 0; j < 32; ++j) {
        const float h = fmaf(z1, w1[32 + j], fmaf(z0, w1[j], b1[j]));
        const float e = __expf(2.0f * h);
        const float t = 1.0f - 2.0f / (e + 1.0f);
        a0 = fmaf(t, w2[2 * j],     a0);
        a1 = fmaf(t, w2[2 * j + 1], a1);
    }
    o0 = a0; o1 = a1;
}

// ---------------------------------------------------------------------------
// ODE integrator + fused decoder.  One thread per trajectory.
// RK4 macro-steps spanning 5 output intervals (H = 5/255, global err ~H^4
// ~= 1.5e-7 << rtol 1e-4), cubic-Hermite dense output at the 5 interior
// grid points (dopri5-style dense output).  x_hat = z @ Wd + bd fused in.
// ---------------------------------------------------------------------------
__global__ __launch_bounds__(256)
void ode_kernel(const float* __restrict__ Wo1, const float* __restrict__ bo1,
                const float* __restrict__ Wo2, const float* __restrict__ bo2,
                const float* __restrict__ Wd,  const float* __restrict__ bd,
                const float* __restrict__ z0_in,
                float* __restrict__ z_traj,    // (T, B, 2)
                float* __restrict__ x_hat)     // (T, B, 3)
{
    __shared__ float sW1[64], sB1[32], sW2[64], sMisc[16]; // bo2[2], Wd[6], bd[3]

    const int tid = threadIdx.x;
    if (tid < 64)  sW1[tid] = Wo1[tid];
    if (tid >= 64 && tid < 128) sW2[tid - 64] = Wo2[tid - 64];
    if (tid >= 128 && tid < 160) sB1[tid - 128] = bo1[tid - 128];
    if (tid >= 160 && tid < 162) sMisc[tid - 160] = bo2[tid - 160];
    if (tid >= 162 && tid < 168) sMisc[tid - 160] = Wd[tid - 162];
    if (tid >= 168 && tid < 171) sMisc[tid - 160] = bd[tid - 168];
    __syncthreads();

    const int b = blockIdx.x * blockDim.x + tid;
    if (b >= B_TOTAL) return;

    const float b20 = sMisc[0], b21 = sMisc[1];
    const float wd00 = sMisc[2], wd01 = sMisc[3], wd02 = sMisc[4];
    const float wd10 = sMisc[5], wd11 = sMisc[6], wd12 = sMisc[7];
    const float bd0 = sMisc[8], bd1 = sMisc[9], bd2 = sMisc[10];

    float zc0 = z0_in[(size_t)b * 2 + 0];
    float zc1 = z0_in[(size_t)b * 2 + 1];

    const float H = 5.0f / 255.0f;   // macro step = 5 output intervals
    float f00, f01;
    vf2(zc0, zc1, sW1, sB1, sW2, b20, b21, f00, f01);

    for (int ms = 0; ms < 51; ++ms) {
        // RK4 macro step (k1 = f00 reused from previous iteration)
        float k20, k21, k30, k31, k40, k41;
        vf2(fmaf(0.5f * H, f00, zc0), fmaf(0.5f * H, f01, zc1),
            sW1, sB1, sW2, b20, b21, k20, k21);
        vf2(fmaf(0.5f * H, k20, zc0), fmaf(0.5f * H, k21, zc1),
            sW1, sB1, sW2, b20, b21, k30, k31);
        vf2(fmaf(H, k30, zc0), fmaf(H, k31, zc1),
            sW1, sB1, sW2, b20, b21, k40, k41);
        const float zn0 = zc0 + H * (1.0f / 6.0f) * (f00 + 2.0f * k20 + 2.0f * k30 + k40);
        const float zn1 = zc1 + H * (1.0f / 6.0f) * (f01 + 2.0f * k21 + 2.0f * k31 + k41);
        float fn0, fn1;
        vf2(zn0, zn1, sW1, sB1, sW2, b20, b21, fn0, fn1);

        // Dense output: cubic Hermite on [zc, zn] at s = j/5, j = 0..4
        #pragma unroll
        for (int j = 0; j < 5; ++j) {
            const float s   = 0.2f * (float)j;
            const float s2  = s * s;
            const float om  = 1.0f - s;
            const float om2 = om * om;
            const float h00 = (1.0f + 2.0f * s) * om2;
            const float h10 = s * om2;
            const float h01 = s2 * (3.0f - 2.0f * s);
            const float h11 = s2 * (s - 1.0f);
            const float p0 = h00 * zc0 + h10 * H * f00 + h01 * zn0 + h11 * H * fn0;
            const float p1 = h00 * zc1 + h10 * H * f01 + h01 * zn1 + h11 * H * fn1;

            const size_t idx = (size_t)(ms * 5 + j) * B_TOTAL + b;
            z_traj[idx * 2 + 0] = p0;
            z_traj[idx * 2 + 1] = p1;
            x_hat[idx * 3 + 0] = fmaf(p0, wd00, fmaf(p1, wd10, bd0));
            x_hat[idx * 3 + 1] = fmaf(p0, wd01, fmaf(p1, wd11, bd1));
            x_hat[idx * 3 + 2] = fmaf(p0, wd02, fmaf(p1, wd12, bd2));
        }

        zc0 = zn0; zc1 = zn1; f00 = fn0; f01 = fn1;
    }

    // Final grid point t = 255 (t = 1.0)
    {
        const size_t idx = (size_t)255 * B_TOTAL + b;
        z_traj[idx * 2 + 0] = zc0;
        z_traj[idx * 2 + 1] = zc1;
        x_hat[idx * 3 + 0] = fmaf(zc0, wd00, fmaf(zc1, wd10, bd0));
        x_hat[idx * 3 + 1] = fmaf(zc0, wd01, fmaf(zc1, wd11, bd1));
        x_hat[idx * 3 + 2] = fmaf(zc0, wd02, fmaf(zc1, wd12, bd2));
    }
}

extern "C" void kernel_launch(void* const* d_in, const int* in_sizes, int n_in,
                              void* d_out, int out_size, void* d_ws, size_t ws_size,
                              hipStream_t stream) {
    const float* x   = (const float*)d_in[0];
    const float* We1 = (const float*)d_in[1];
    const float* be1 = (const float*)d_in[2];
    const float* We2 = (const float*)d_in[3];
    const float* be2 = (const float*)d_in[4];
    const float* We3 = (const float*)d_in[5];
    const float* be3 = (const float*)d_in[6];
    const float* Wo1 = (const float*)d_in[7];
    const float* bo1 = (const float*)d_in[8];
    const float* Wo2 = (const float*)d_in[9];
    const float* bo2 = (const float*)d_in[10];
    const float* Wd  = (const float*)d_in[11];
    const float* bd  = (const float*)d_in[12];

    float* out    = (float*)d_out;
    float* z0     = out;                                   // (B, 2)
    float* z_traj = out + (size_t)B_TOTAL * 2;             // (T, B, 2)
    float* x_hat  = z_traj + (size_t)T_STEPS * B_TOTAL * 2;// (T, B, 3)

    encoder_kernel<<<B_TOTAL / 16, 32, 0, stream>>>(
        x, We1, be1, We2, be2, We3, be3, z0);
    ode_kernel<<<B_TOTAL / 256, 256, 0, stream>>>(
        Wo1, bo1, Wo2, bo2, Wd, bd, z0, z_traj, x_hat);
}